// UpsampleLayer2D_6734508720545
// MI455X (gfx1250) — compile-verified
//
#include <hip/hip_runtime.h>

// UpsampleLayer2D for MI455X (gfx1250): pure streaming quadrant pixel-shuffle.
//   out[b, r + jr*R, col + jc*C, c] = x[b, r, col, 4c + 2*jr + jc]
// 655 MB mandatory HBM traffic, 0 FLOPs -> bandwidth bound (~28 us @ 23.3 TB/s).
//
// Launch geometry encodes the tensor geometry so there is ZERO per-thread
// division and no bounds check:
//   grid = (10, R=200, B=32), block = 320 threads (10 wave32 waves)
//   t = blockIdx.x*320 + threadIdx.x  in [0, 3200)  == col*16 + c
// Per thread:
//   - 1x global_load_b128 th:TH_LOAD_NT   (512 B contiguous per wave)
//   - 4x global_store_b32 th:TH_STORE_NT  (128 B contiguous per wave each,
//     quadrant hops are compile-time constants folded into 24-bit IOFFSET)
// All (b, r)-dependent addressing is wave-uniform -> SALU.

typedef __attribute__((ext_vector_type(4))) float v4f;

namespace {

constexpr int kB  = 32;
constexpr int kR  = 200;
constexpr int kC  = 200;
constexpr int kCo = 16;                       // IN_CH / 4
constexpr int kRowV4   = kC * kCo;            // 3,200 float4s per input pixel-row
constexpr int kThreads = 320;                 // 10 waves; 3200 / 320 = 10 blocks/row
constexpr int kBlocksX = kRowV4 / kThreads;   // 10

// Output hops (in float elements)
constexpr int kColHop = kC * kCo;             // right quadrant:      3,200
constexpr int kRowHop = kR * (2 * kC) * kCo;  // bottom quadrant: 1,280,000
constexpr int kOutRow = (2 * kC) * kCo;       // one output row:      6,400

__global__ __launch_bounds__(kThreads) void upsample2d_pixelshuffle(
    const v4f* __restrict__ x4, float* __restrict__ out) {
  const int t = blockIdx.x * kThreads + threadIdx.x;  // col*16 + c, in [0, 3200)
  const int r = blockIdx.y;
  const int b = blockIdx.z;

  // Contiguous 16B load: x[b, r, col, 4c .. 4c+3]  (the 4 quadrant values)
  const v4f* src = x4 + (b * kR + r) * kRowV4 + t;
  v4f v = __builtin_nontemporal_load(src);

  // Base output element: (b, r, col, c) in the [B, 2R, 2C, Co] tensor.
  // Left half of output row r is laid out exactly as col*Co + c == t.
  float* o = out + (b * (2 * kR) + r) * kOutRow + t;

  __builtin_nontemporal_store(v.x, o);                      // top-left     (jr=0,jc=0)
  __builtin_nontemporal_store(v.y, o + kColHop);            // top-right    (jr=0,jc=1)
  __builtin_nontemporal_store(v.z, o + kRowHop);            // bottom-left  (jr=1,jc=0)
  __builtin_nontemporal_store(v.w, o + kRowHop + kColHop);  // bottom-right (jr=1,jc=1)
}

}  // namespace

extern "C" void kernel_launch(void* const* d_in, const int* in_sizes, int n_in,
                              void* d_out, int out_size, void* d_ws, size_t ws_size,
                              hipStream_t stream) {
  (void)in_sizes; (void)n_in; (void)out_size; (void)d_ws; (void)ws_size;
  const v4f* x4 = (const v4f*)d_in[0];
  float* out = (float*)d_out;

  dim3 grid(kBlocksX, kR, kB);  // (10, 200, 32) -> exact cover, no tail
  upsample2d_pixelshuffle<<<grid, kThreads, 0, stream>>>(x4, out);
}